// Sampler_32942399161106
// MI455X (gfx1250) — compile-verified
//
#include <hip/hip_runtime.h>
#include <hip/hip_bf16.h>

#define NSEQ   256
#define VOCABN 128256
#define HISTN  2048
#define HASHN  4096
#define HMASK  (HASHN - 1)
#define BINS   16384
#define KEYSHIFT 18            // keep top 14 bits of monotonic key
#define CMAX   4096
#define TPB    256
#define KCAP   2048            // top_k < 2000 <= 2048
#define HEMPTY 0xFFFFFFFFu

#ifndef __has_builtin
#define __has_builtin(x) 0
#endif

typedef int v4i __attribute__((ext_vector_type(4)));

// ---------- helpers ----------
__device__ __forceinline__ unsigned int fmono(float f) {
  unsigned int u = __float_as_uint(f);
  return (u & 0x80000000u) ? ~u : (u | 0x80000000u);
}
__device__ __forceinline__ float funmono(unsigned int u) {
  unsigned int b = (u & 0x80000000u) ? (u ^ 0x80000000u) : ~u;
  return __uint_as_float(b);
}

__device__ __forceinline__ unsigned int hprobe(const unsigned int* hkey,
                                               const unsigned int* hcnt,
                                               unsigned int tok) {
  unsigned int h = (tok * 2654435761u) & HMASK;
  while (true) {
    unsigned int k = hkey[h];
    if (k == tok) return hcnt[h];
    if (k == HEMPTY) return 0u;
    h = (h + 1u) & HMASK;
  }
}

// Mirrors the reference's float op ordering exactly.
__device__ __forceinline__ float penal(float x, unsigned int cnt,
                                       float fp, float pp, float rp, float tr) {
  if (cnt > 0u) {
    x = x - fp * (float)cnt;
    x = x - pp;
    if (x > 0.0f) x = x + x * (1.0f / rp - 1.0f);
    if (x < 0.0f) x = x + x * (rp - 1.0f);
  }
  return x / tr;
}

// ---------- kernel 1: fill output with -inf ----------
__global__ void __launch_bounds__(TPB) fill_neg_inf(float4* __restrict__ out4, int n4) {
  const float NI = __int_as_float(0xFF800000);
  float4 v; v.x = NI; v.y = NI; v.z = NI; v.w = NI;
  for (int i = blockIdx.x * blockDim.x + threadIdx.x; i < n4; i += gridDim.x * blockDim.x)
    out4[i] = v;
}

// ---------- kernel 2: one workgroup per sequence ----------
__global__ void __launch_bounds__(TPB) sampler_select(
    const float* __restrict__ logits, const int* __restrict__ hist,
    const float* __restrict__ pres, const float* __restrict__ freq,
    const float* __restrict__ rep,  const float* __restrict__ temp,
    const float* __restrict__ topp, const float* __restrict__ topa,
    const int*  __restrict__ topk,  float* __restrict__ out)
{
  __shared__ unsigned int hkey[HASHN];                 // 16 KB
  __shared__ unsigned int hcnt[HASHN];                 // 16 KB
  __shared__ __align__(16) unsigned int histo[BINS];   // 64 KB (reused for cand+ps)
  __shared__ __align__(16) int ldsTok[HISTN];          // 8 KB
  __shared__ unsigned int part[TPB];
  __shared__ unsigned int sufx[TPB];
  __shared__ float red[TPB];
  __shared__ float fscan[TPB];
  __shared__ int sh_selB2;
  __shared__ unsigned int sh_ncand;
  __shared__ int sh_nkeep;

  unsigned long long* cand = reinterpret_cast<unsigned long long*>(histo); // 4096 u64 = 32 KB
  float* ps = reinterpret_cast<float*>(histo + 2 * CMAX);                  // 2048 f32 = 8 KB

  const int row = blockIdx.x;
  const int tid = threadIdx.x;
  const float fp = freq[row], pp = pres[row], rp = rep[row], tr = temp[row];
  const float tp = topp[row], ta = topa[row];
  int Ksel = topk[row];
  if (Ksel > KCAP) Ksel = KCAP;

  for (int i = tid; i < HASHN; i += TPB) { hkey[i] = HEMPTY; hcnt[i] = 0u; }
  for (int i = tid; i < BINS; i += TPB) histo[i] = 0u;
  if (tid == 0) { sh_selB2 = 0; sh_ncand = 0u; sh_nkeep = 0; }

  // ---- stage history tokens into LDS (gfx1250 async global->LDS path) ----
  const int* gh = hist + (size_t)row * HISTN;
#if __has_builtin(__builtin_amdgcn_global_load_async_to_lds_b128)
  {
    auto* gp = (__attribute__((address_space(1))) v4i*)(void*)gh;
    auto* lp = (__attribute__((address_space(3))) v4i*)(&ldsTok[0]);
    // 2048 ints = 512 x int4; 256 lanes x 2 issues
    __builtin_amdgcn_global_load_async_to_lds_b128(gp + tid, lp + tid, 0, 0);
    __builtin_amdgcn_global_load_async_to_lds_b128(gp + TPB + tid, lp + TPB + tid, 0, 0);
#if __has_builtin(__builtin_amdgcn_s_wait_asynccnt)
    __builtin_amdgcn_s_wait_asynccnt(0);
#else
    asm volatile("s_wait_asynccnt 0" ::: "memory");
#endif
  }
#else
  for (int t = tid; t < HISTN; t += TPB) ldsTok[t] = gh[t];
#endif
  __syncthreads();

  // ---- build token->count hash table in LDS ----
  for (int t = tid; t < HISTN; t += TPB) {
    unsigned int tok = (unsigned int)ldsTok[t];
    unsigned int h = (tok * 2654435761u) & HMASK;
    while (true) {
      unsigned int prev = atomicCAS(&hkey[h], HEMPTY, tok);
      if (prev == HEMPTY || prev == tok) { atomicAdd(&hcnt[h], 1u); break; }
      h = (h + 1u) & HMASK;
    }
  }
  __syncthreads();

  // ---- pass 1: histogram of monotonic keys ----
  const float4* lrow4 = (const float4*)(logits + (size_t)row * VOCABN);
  for (int i = tid; i < VOCABN / 4; i += TPB) {
    __builtin_prefetch(lrow4 + i + TPB * 8, 0, 0);
    float4 l4 = lrow4[i];
    int v0 = i * 4;
    float x0 = penal(l4.x, hprobe(hkey, hcnt, (unsigned)(v0 + 0)), fp, pp, rp, tr);
    float x1 = penal(l4.y, hprobe(hkey, hcnt, (unsigned)(v0 + 1)), fp, pp, rp, tr);
    float x2 = penal(l4.z, hprobe(hkey, hcnt, (unsigned)(v0 + 2)), fp, pp, rp, tr);
    float x3 = penal(l4.w, hprobe(hkey, hcnt, (unsigned)(v0 + 3)), fp, pp, rp, tr);
    atomicAdd(&histo[fmono(x0) >> KEYSHIFT], 1u);
    atomicAdd(&histo[fmono(x1) >> KEYSHIFT], 1u);
    atomicAdd(&histo[fmono(x2) >> KEYSHIFT], 1u);
    atomicAdd(&histo[fmono(x3) >> KEYSHIFT], 1u);
  }
  __syncthreads();

  // ---- find bin threshold B: largest bin with count(key>=B) >= top_k ----
  {
    const int SEG = BINS / TPB; // 64
    int b0 = tid * SEG;
    unsigned int s = 0;
    for (int j = 0; j < SEG; ++j) s += histo[b0 + j];
    part[tid] = s;
    __syncthreads();
    if (tid == 0) {
      unsigned int run = 0;
      for (int q = TPB - 1; q >= 0; --q) { sufx[q] = run; run += part[q]; }
    }
    __syncthreads();
    unsigned int run = sufx[tid];
    for (int j = SEG - 1; j >= 0; --j) {
      unsigned int c = histo[b0 + j];
      if (c > 0u && run < (unsigned)Ksel && run + c >= (unsigned)Ksel) {
        // unique writer; back off one bin if it would overflow candidate buffer
        sh_selB2 = (run + c > (unsigned)CMAX) ? (b0 + j + 1) : (b0 + j);
      }
      run += c;
    }
    __syncthreads();
  }
  const int selB2 = sh_selB2;
  __syncthreads();   // everyone done with histo before it becomes cand[]

  // ---- pass 2: collect candidates (value<<32 | ~index) ----
  for (int i = tid; i < VOCABN / 4; i += TPB) {
    __builtin_prefetch(lrow4 + i + TPB * 8, 0, 0);
    float4 l4 = lrow4[i];
    int v0 = i * 4;
#pragma unroll
    for (int j = 0; j < 4; ++j) {
      float lx = (j == 0) ? l4.x : (j == 1) ? l4.y : (j == 2) ? l4.z : l4.w;
      float x = penal(lx, hprobe(hkey, hcnt, (unsigned)(v0 + j)), fp, pp, rp, tr);
      unsigned int mk = fmono(x);
      if ((int)(mk >> KEYSHIFT) >= selB2) {
        unsigned int slot = atomicAdd(&sh_ncand, 1u);
        if (slot < CMAX)
          cand[slot] = ((unsigned long long)mk << 32) | (unsigned int)~((unsigned int)(v0 + j));
      }
    }
  }
  __syncthreads();
  int ncand = (int)(sh_ncand < (unsigned)CMAX ? sh_ncand : (unsigned)CMAX);
  for (int i = ncand + tid; i < CMAX; i += TPB) cand[i] = 0ull;

  // ---- bitonic sort, descending (stable wrt reference tie-break via ~index) ----
  for (unsigned int k = 2; k <= CMAX; k <<= 1) {
    for (unsigned int j = k >> 1; j > 0; j >>= 1) {
      __syncthreads();
      for (unsigned int i = tid; i < CMAX; i += TPB) {
        unsigned int ixj = i ^ j;
        if (ixj > i) {
          unsigned long long a = cand[i], b = cand[ixj];
          bool desc = ((i & k) == 0u);
          if (desc ? (a < b) : (a > b)) { cand[i] = b; cand[ixj] = a; }
        }
      }
    }
  }
  __syncthreads();

  // ---- softmax over first K, cumsum, find n_keep (prefix property) ----
  int K = Ksel < ncand ? Ksel : ncand;
  if (K > 0) {
    float m = funmono((unsigned int)(cand[0] >> 32));
    float lsum = 0.0f;
    for (int i = tid; i < K; i += TPB) {
      float e = expf(funmono((unsigned int)(cand[i] >> 32)) - m);
      ps[i] = e;
      lsum += e;
    }
    red[tid] = lsum;
    for (int off = TPB / 2; off > 0; off >>= 1) {
      __syncthreads();
      if (tid < off) red[tid] += red[tid + off];
    }
    __syncthreads();
    float Z = red[0];
    if (tid == 0) sh_nkeep = K;
    __syncthreads();

    const int CH = KCAP / TPB; // 8 per thread, contiguous
    int base = tid * CH;
    float c8[CH];
    float ssum = 0.0f;
#pragma unroll
    for (int j = 0; j < CH; ++j) {
      int i = base + j;
      float p = (i < K) ? (ps[i] / Z) : 0.0f;
      c8[j] = p;
      ssum += p;
    }
    red[tid] = ssum;
    __syncthreads();
    if (tid == 0) {
      float r = 0.0f;
      for (int q = 0; q < TPB; ++q) { fscan[q] = r; r += red[q]; }
    }
    __syncthreads();
    float p0 = 1.0f / Z;               // exp(0)/Z, matches reference probs[:,0]
    float thr = p0 * p0 * ta;
    float runp = fscan[tid];
#pragma unroll
    for (int j = 0; j < CH; ++j) {
      int i = base + j;
      if (i < K) {
        runp += c8[j];
        if (c8[j] < thr || runp > tp) atomicMin(&sh_nkeep, i);
      }
    }
    __syncthreads();
    if (tid == 0 && row == 0 && sh_nkeep < 1) sh_nkeep = 1; // mask[0,0]=False quirk
    __syncthreads();

    // ---- scatter kept values (everything else already -inf) ----
    int nkeep = sh_nkeep;
    float* orow = out + (size_t)row * VOCABN;
    for (int i = tid; i < nkeep; i += TPB) {
      unsigned long long c = cand[i];
      unsigned int v = ~((unsigned int)c);
      orow[v] = funmono((unsigned int)(c >> 32));
    }
  }
}

extern "C" void kernel_launch(void* const* d_in, const int* in_sizes, int n_in,
                              void* d_out, int out_size, void* d_ws, size_t ws_size,
                              hipStream_t stream) {
  (void)in_sizes; (void)n_in; (void)d_ws; (void)ws_size; (void)out_size;
  const float* logits = (const float*)d_in[0];
  const int*   hist   = (const int*)d_in[1];
  const float* pres   = (const float*)d_in[2];
  const float* freq   = (const float*)d_in[3];
  const float* rep    = (const float*)d_in[4];
  const float* temp   = (const float*)d_in[5];
  const float* topp   = (const float*)d_in[6];
  const float* topa   = (const float*)d_in[7];
  const int*   topk   = (const int*)d_in[8];
  float* out = (float*)d_out;

  const int n4 = (NSEQ * VOCABN) / 4;
  fill_neg_inf<<<dim3(2048), dim3(TPB), 0, stream>>>((float4*)out, n4);
  sampler_select<<<dim3(NSEQ), dim3(TPB), 0, stream>>>(
      logits, hist, pres, freq, rep, temp, topp, topa, topk, out);
}